// CapsuleLayer_24876450579071
// MI455X (gfx1250) — compile-verified
//
#include <hip/hip_runtime.h>
#include <hip/hip_bf16.h>

typedef float v8f __attribute__((ext_vector_type(8)));
typedef float v2f __attribute__((ext_vector_type(2)));

#define B_   64
#define J_   2048
#define DI_  16
#define KC_  32
#define DO_  32
#define NC_  (KC_*DO_)     // 1024
#define TB_  16            // b per workgroup
#define TJ_  32            // j per workgroup
#define JT_  (J_/TJ_)      // 64 j-tiles
#define THREADS_ 256
#define EPS_ 1e-7f

// workspace layout (in floats)
#define SPART_SZ (JT_*B_*NC_)   // 4,194,304  (16.8 MB) deterministic partial s
#define SRED_SZ  (B_*NC_)       // 65,536     reduced s
#define VA_SZ    (B_*NC_)       // 65,536     accumulated v (v0, then v0+v1)

// ---------------------------------------------------------------------------
// Fused pass: recompute inputs_hat tiles with fp32 WMMA, do routing math in
// registers, write deterministic per-(j-tile) partial sums of s.
// MODE 0: c uniform (= softmax of zeros = 1/32), no va needed.
// MODE 1: logits = va . ihat (dot over o), softmax over k, s += c * ihat.
// GEMM orientation: M=(k,o) rows = W^T, N=b cols = X^T, K=Di=16 (4 steps of 4).
// ---------------------------------------------------------------------------
template <int MODE>
__global__ __launch_bounds__(THREADS_)
void caps_pass(const float* __restrict__ X, const float* __restrict__ W,
               const float* __restrict__ va, float* __restrict__ spart)
{
    __shared__ float Xs[2][TB_][18];     // padded: stride 18 -> conflict-free, 8B aligned pairs
    __shared__ float logit_s[TB_][33];   // padded: stride 33 -> conflict-free

    const int jt = blockIdx.x;
    const int bt = blockIdx.y;
    const int b0 = bt * TB_;
    const int tid  = threadIdx.x;
    const int w    = tid >> 5;           // wave 0..7 -> k range [4w, 4w+4)
    const int lane = tid & 31;
    const int ln   = lane & 15;          // C layout: column b = ln for both halves
    const int h    = lane >> 4;          // half: row offset +8, A/B K offset +2
    const int bg   = b0 + ln;            // global b for this lane's columns
    const int j0   = jt * TJ_;

    // Preload va values matching this lane's C-layout positions (MODE 1 only).
    float var_[64];
    if (MODE) {
        #pragma unroll
        for (int t = 0; t < 8; ++t) {
            const int k  = 4*w + (t >> 1);
            const int ob = (t & 1) * 16;
            #pragma unroll
            for (int r = 0; r < 8; ++r)
                var_[t*8 + r] = va[(bg*KC_ + k)*DO_ + ob + r + 8*h];
        }
    }

    v8f sacc[8];
    #pragma unroll
    for (int t = 0; t < 8; ++t) sacc[t] = (v8f){0,0,0,0,0,0,0,0};

    // stage X for first j
    {
        const int m = tid >> 4, i = tid & 15;
        Xs[0][m][i] = X[((size_t)(b0 + m)*J_ + j0)*DI_ + i];
    }

    for (int jj = 0; jj < TJ_; ++jj) {
        const int j   = j0 + jj;
        const int buf = jj & 1;
        __syncthreads();   // Xs[buf] ready; prior logit reads complete

        // stage next j's X into the other buffer (issue load early)
        float xnext = 0.f;
        const int sm = tid >> 4, si = tid & 15;
        if (jj + 1 < TJ_)
            xnext = X[((size_t)(b0 + sm)*J_ + (j + 1))*DI_ + si];

        // B fragments (X^T: Di x b), shared across all 8 M-tiles.
        // B 4x16 layout: VGPR0 -> K = 4s+2h, VGPR1 -> K = 4s+2h+1, col N = ln.
        v2f bf[4];
        #pragma unroll
        for (int s = 0; s < 4; ++s) {
            const int i0 = 4*s + 2*h;
            bf[s].x = Xs[buf][ln][i0];
            bf[s].y = Xs[buf][ln][i0 + 1];
        }

        if (jj + 1 < TJ_) Xs[buf ^ 1][sm][si] = xnext;

        // GEMM: ihat tile = W_j^T (1024 x 16) x X_j^T (16 x 16), fp32 WMMA.
        const float* Wj = W + (size_t)j * (KC_*DI_*DO_);
        v8f acc[8];
        #pragma unroll
        for (int t = 0; t < 8; ++t) {
            const int k  = 4*w + (t >> 1);
            const int ob = (t & 1) * 16;
            // A 16x4 layout: lane row M = ln -> o = ob+ln; K rows 4s+2h, 4s+2h+1.
            const float* Wp = Wj + k*(DI_*DO_) + (2*h)*DO_ + ob + ln;
            v8f a = (v8f){0,0,0,0,0,0,0,0};
            #pragma unroll
            for (int s = 0; s < 4; ++s) {
                v2f af;
                af.x = Wp[(4*s    )*DO_];
                af.y = Wp[(4*s + 1)*DO_];
                a = __builtin_amdgcn_wmma_f32_16x16x4_f32(
                        false, af, false, bf[s], (short)0, a, false, false);
            }
            acc[t] = a;
        }

        // Routing coefficients
        float c4[4];
        if (MODE) {
            // logits[b,k] = sum_o va[b,k,o]*ihat[b,k,o]; o lives in-lane (16 of 32)
            #pragma unroll
            for (int q = 0; q < 4; ++q) {
                float p = 0.f;
                #pragma unroll
                for (int r = 0; r < 8; ++r)
                    p += var_[(2*q)*8 + r]   * acc[2*q][r]
                       + var_[(2*q+1)*8 + r] * acc[2*q+1][r];
                p += __shfl_xor(p, 16);          // combine the two o-halves
                if (h == 0) logit_s[ln][4*w + q] = p;
            }
            __syncthreads();
            // per-wave softmax over k for b = ln (redundant across halves/waves)
            float mx = -1e30f;
            #pragma unroll
            for (int kk = 0; kk < KC_; ++kk) mx = fmaxf(mx, logit_s[ln][kk]);
            float Z = 0.f;
            #pragma unroll
            for (int kk = 0; kk < KC_; ++kk) Z += __expf(logit_s[ln][kk] - mx);
            const float rZ = 1.f / Z;
            #pragma unroll
            for (int q = 0; q < 4; ++q)
                c4[q] = __expf(logit_s[ln][4*w + q] - mx) * rZ;
        } else {
            c4[0] = c4[1] = c4[2] = c4[3] = 1.f;   // 1/32 folded into epilogue
        }

        // s accumulation (registers, deterministic order over j)
        #pragma unroll
        for (int t = 0; t < 8; ++t) {
            const float c = c4[t >> 1];
            #pragma unroll
            for (int r = 0; r < 8; ++r) sacc[t][r] += c * acc[t][r];
        }
    }

    // write deterministic partials for this j-tile
    const float scale = MODE ? 1.f : (1.f / (float)KC_);
    float* dst = spart + (size_t)jt * (B_*NC_);
    #pragma unroll
    for (int t = 0; t < 8; ++t) {
        const int k  = 4*w + (t >> 1);
        const int ob = (t & 1) * 16;
        #pragma unroll
        for (int r = 0; r < 8; ++r) {
            const int o = ob + r + 8*h;
            dst[(bg*KC_ + k)*DO_ + o] = sacc[t][r] * scale;
        }
    }
}

// fixed-order reduction over j-tiles -> deterministic s
__global__ __launch_bounds__(256)
void caps_reduce(const float* __restrict__ spart, float* __restrict__ sred)
{
    const int idx = blockIdx.x * 256 + threadIdx.x;   // 65536 threads
    float acc = 0.f;
    #pragma unroll 8
    for (int jt = 0; jt < JT_; ++jt)
        acc += spart[(size_t)jt * (B_*NC_) + idx];
    sred[idx] = acc;
}

// v = squash(s); va = v (first) or va += v; final writes d_out
__global__ __launch_bounds__(256)
void caps_squash(const float* __restrict__ sred, float* __restrict__ va,
                 float* __restrict__ out, int first, int final_)
{
    const int idx = blockIdx.x * 256 + threadIdx.x;   // 2048 threads: (b,k)
    const float* sp = sred + (size_t)idx * DO_;
    float v[DO_];
    float s2 = 0.f;
    #pragma unroll
    for (int o = 0; o < DO_; ++o) { const float x = sp[o]; v[o] = x; s2 += x*x; }
    const float scale = s2 / ((1.f + s2) * sqrtf(s2 + EPS_));
    if (final_) {
        float* op = out + (size_t)idx * DO_;
        #pragma unroll
        for (int o = 0; o < DO_; ++o) op[o] = scale * v[o];
    } else {
        float* vp = va + (size_t)idx * DO_;
        if (first) {
            #pragma unroll
            for (int o = 0; o < DO_; ++o) vp[o] = scale * v[o];
        } else {
            #pragma unroll
            for (int o = 0; o < DO_; ++o) vp[o] += scale * v[o];
        }
    }
}

extern "C" void kernel_launch(void* const* d_in, const int* in_sizes, int n_in,
                              void* d_out, int out_size, void* d_ws, size_t ws_size,
                              hipStream_t stream)
{
    const float* X = (const float*)d_in[0];   // [64, 2048, 16]
    const float* W = (const float*)d_in[1];   // [2048, 32, 16, 32]
    float* out = (float*)d_out;               // [64, 32, 32]

    float* ws    = (float*)d_ws;
    float* spart = ws;                        // SPART_SZ
    float* sred  = ws + SPART_SZ;             // SRED_SZ
    float* va    = ws + SPART_SZ + SRED_SZ;   // VA_SZ

    const dim3 pg(JT_, B_/TB_);               // (64, 4)

    // routing iteration 0: c uniform
    caps_pass<0><<<pg, THREADS_, 0, stream>>>(X, W, va, spart);
    caps_reduce<<<SRED_SZ/256, 256, 0, stream>>>(spart, sred);
    caps_squash<<<(B_*KC_)/256, 256, 0, stream>>>(sred, va, out, 1, 0);  // va = v0

    // routing iteration 1: logits = v0 . ihat
    caps_pass<1><<<pg, THREADS_, 0, stream>>>(X, W, va, spart);
    caps_reduce<<<SRED_SZ/256, 256, 0, stream>>>(spart, sred);
    caps_squash<<<(B_*KC_)/256, 256, 0, stream>>>(sred, va, out, 0, 0);  // va = v0+v1

    // routing iteration 2: logits = (v0+v1) . ihat ; output v2
    caps_pass<1><<<pg, THREADS_, 0, stream>>>(X, W, va, spart);
    caps_reduce<<<SRED_SZ/256, 256, 0, stream>>>(spart, sred);
    caps_squash<<<(B_*KC_)/256, 256, 0, stream>>>(sred, va, out, 0, 1);  // d_out = v2
}